// DropBlock_49624052138010
// MI455X (gfx1250) — compile-verified
//
#include <hip/hip_runtime.h>

// DropBlock on (32, 256, 56, 56) fp32.
// Kernel 1: build scale-premultiplied 56x56 block mask into d_ws (single WG, LDS).
// Kernel 2: pure-streaming out = x * mask (mask cached in registers, NT hints).

#define FEAT_H     56
#define FEAT_W     56
#define HW         (FEAT_H * FEAT_W)   // 3136
#define HW4        (HW / 4)            // 784 float4 per plane
#define BLOCK_SZ   7
#define NTHREADS   256
#define PLANES_PER_BLOCK 8

typedef float f4 __attribute__((ext_vector_type(4)));

// ---------------------------------------------------------------------------
// Kernel 1: seeds -> 7x7 backward max dilation -> block mask -> premultiply by
// scale = HW / sum(mask). Sum is exact (0/1 summands) so ds_add_f32 atomic
// ordering cannot change the result -> deterministic.
// ---------------------------------------------------------------------------
__global__ __launch_bounds__(1024) void dropblock_mask_kernel(
    const float* __restrict__ u, float* __restrict__ mscaled) {
  const float GAMMA =
      (float)((0.1 / (7.0 * 7.0)) * ((56.0 * 56.0) / (50.0 * 50.0)));  // 0.00256

  __shared__ float seed[HW];
  __shared__ float bm[HW];
  __shared__ float ssum;

  const int tid = threadIdx.x;
  const int nthr = blockDim.x;
  if (tid == 0) ssum = 0.0f;

  for (int p = tid; p < HW; p += nthr)
    seed[p] = (u[p] < GAMMA) ? 1.0f : 0.0f;
  __syncthreads();

  float local = 0.0f;
  for (int p = tid; p < HW; p += nthr) {
    const int i = p / FEAT_W;
    const int j = p - i * FEAT_W;
    int a0 = i - (BLOCK_SZ - 1); if (a0 < 0) a0 = 0;
    int b0 = j - (BLOCK_SZ - 1); if (b0 < 0) b0 = 0;
    float d = 0.0f;
    for (int a = a0; a <= i; ++a) {
      const float* row = &seed[a * FEAT_W];
      for (int b = b0; b <= j; ++b) d = fmaxf(d, row[b]);
    }
    const float m = 1.0f - d;   // 0 inside dropped blocks, else 1
    bm[p] = m;
    local += m;
  }
  atomicAdd(&ssum, local);      // ds_add_f32, exact integer-valued sum
  __syncthreads();

  const float scale = (float)HW / ssum;
  for (int p = tid; p < HW; p += nthr)
    mscaled[p] = bm[p] * scale;
}

// ---------------------------------------------------------------------------
// Kernel 2: out = x * mscaled[hw]. One 8-wave block per PLANES_PER_BLOCK
// (N*C) planes. Mask float4s are loaded once into registers and reused across
// planes. x/out are streamed with non-temporal b128 accesses (touched once;
// keep 192MB L2 clean); next plane is prefetched (global_prefetch_b8).
// ---------------------------------------------------------------------------
__global__ __launch_bounds__(NTHREADS) void dropblock_apply_kernel(
    const f4* __restrict__ x, const f4* __restrict__ m, f4* __restrict__ out,
    int nplanes) {
  const int tid = threadIdx.x;
  const long long plane0 = (long long)blockIdx.x * PLANES_PER_BLOCK;

  // 784 = 3*256 + 16: threads 0..255 cache 3 mask float4s, threads 0..15 a 4th.
  const f4 m0 = m[tid];
  const f4 m1 = m[tid + 256];
  const f4 m2 = m[tid + 512];
  f4 m3 = (f4)(0.0f);
  if (tid < HW4 - 768) m3 = m[tid + 768];

  for (int p = 0; p < PLANES_PER_BLOCK; ++p) {
    const long long plane = plane0 + p;
    if (plane >= nplanes) break;
    const f4* xp = x + plane * HW4;
    f4*       op = out + plane * HW4;

    if (p + 1 < PLANES_PER_BLOCK && plane + 1 < nplanes) {
      // Pull the next plane toward the caches while this one streams.
      __builtin_prefetch((const void*)(xp + HW4 + tid), 0, 0);
      __builtin_prefetch((const void*)(xp + HW4 + 512 + tid), 0, 0);
    }

    const f4 a0 = __builtin_nontemporal_load(xp + tid);
    const f4 a1 = __builtin_nontemporal_load(xp + tid + 256);
    const f4 a2 = __builtin_nontemporal_load(xp + tid + 512);
    __builtin_nontemporal_store(a0 * m0, op + tid);
    __builtin_nontemporal_store(a1 * m1, op + tid + 256);
    __builtin_nontemporal_store(a2 * m2, op + tid + 512);
    if (tid < HW4 - 768) {
      const f4 a3 = __builtin_nontemporal_load(xp + tid + 768);
      __builtin_nontemporal_store(a3 * m3, op + tid + 768);
    }
  }
}

// ---------------------------------------------------------------------------
extern "C" void kernel_launch(void* const* d_in, const int* in_sizes, int n_in,
                              void* d_out, int out_size, void* d_ws, size_t ws_size,
                              hipStream_t stream) {
  const float* x = (const float*)d_in[0];   // (32,256,56,56) fp32
  const float* u = (const float*)d_in[1];   // (56,56) fp32
  float* out = (float*)d_out;
  float* mscaled = (float*)d_ws;            // 3136 floats of scratch

  dropblock_mask_kernel<<<1, 1024, 0, stream>>>(u, mscaled);

  const int nplanes = in_sizes[0] / HW;     // 32*256 = 8192
  const int nblocks = (nplanes + PLANES_PER_BLOCK - 1) / PLANES_PER_BLOCK;
  dropblock_apply_kernel<<<nblocks, NTHREADS, 0, stream>>>(
      (const f4*)x, (const f4*)mscaled, (f4*)out, nplanes);
}